// MEModule_59545426592351
// MI455X (gfx1250) — compile-verified
//
#include <hip/hip_runtime.h>
#include <hip/hip_bf16.h>

typedef __attribute__((ext_vector_type(2))) float v2f;
typedef __attribute__((ext_vector_type(4))) float v4f;
typedef __attribute__((ext_vector_type(8))) float v8f;

// geometry
#define N_D    24
#define N_OUT  128
#define NT1    10      // GEMM1 N-tiles: 5 'a' tiles + 5 'g' tiles (72 cols padded to 80 each)
#define K1     6       // 24 / 4
#define KP1    3       // k-step pairs, GEMM1
#define NT2    8       // 128 / 16
#define K2     18      // 72 / 4
#define KP2    9       // k-step pairs, GEMM2
#define SSTRIDE 76     // padded LDS row stride (dwords) for the stacked S tile

// LDS layout (dword offsets). B-fragments stored as k-PAIRS: 4 dwords/lane,
// 16B aligned, so one ds_load_b128 yields two WMMA B operands.
#define WB1_OFF 0
#define WB1_SZ  (NT1 * KP1 * 32 * 4)         // 3840
#define WB2_OFF (WB1_OFF + WB1_SZ)
#define WB2_SZ  (NT2 * KP2 * 32 * 4)         // 9216
#define BA_OFF  (WB2_OFF + WB2_SZ)           // 13056
#define BG_OFF  (BA_OFF + 80)
#define BL_OFF  (BG_OFF + 80)
#define S_OFF   (BL_OFF + 128)               // 13344
#define S_PER_WAVE (16 * SSTRIDE)            // 1216
#define LDS_DWORDS (S_OFF + 8 * S_PER_WAVE)  // 23072 dwords = 92,288 B

__global__ __launch_bounds__(256) void me_glu_kernel(
    const float* __restrict__ rbf, const float* __restrict__ h,
    const int*   __restrict__ idx_s, const int* __restrict__ idx_t,
    const float* __restrict__ w1, const float* __restrict__ b1,
    const float* __restrict__ w2, const float* __restrict__ b2,
    const float* __restrict__ wl, const float* __restrict__ bl,
    float* __restrict__ out, int n_edges)
{
  extern __shared__ float lds[];
  const int tid  = threadIdx.x;
  const int lane = tid & 31;
  const int wave = tid >> 5;
  const bool hi  = (lane >= 16);
  const int  lp  = lane & 15;

  // ---------------- build fragment-swizzled weights in LDS (once per block) -------------
  // GEMM1 B: Wcat[24 x 160]; tiles 0..4 = w1 (cols 0..71 real), tiles 5..9 = w2.
  for (int i = tid; i < WB1_SZ; i += 256) {
    int j2   = i & 3;                              // dword within 16B pair-slot
    int ln   = (i >> 2) & 31;
    int rest = i >> 7;
    int kp   = rest % KP1;
    int nt   = rest / KP1;
    int kk   = 2 * kp + (j2 >> 1);
    int j    = j2 & 1;
    int r = 4 * kk + ((ln < 16) ? j : 2 + j);      // K row within 24
    int c = 16 * (nt % 5) + (ln & 15);             // column within 80-wide (padded) part
    float v = 0.0f;
    if (c < 72) {
      int m = c / 24, f = c % 24;
      const float* w = (nt < 5) ? w1 : w2;         // [3,24,24] : (m,d,f)
      v = w[(m * 24 + r) * 24 + f];
    }
    lds[WB1_OFF + i] = v;
  }
  // GEMM2 B: wl[72 x 128]
  for (int i = tid; i < WB2_SZ; i += 256) {
    int j2   = i & 3;
    int ln   = (i >> 2) & 31;
    int rest = i >> 7;
    int kp   = rest % KP2;
    int nt   = rest / KP2;
    int kk   = 2 * kp + (j2 >> 1);
    int j    = j2 & 1;
    int r = 4 * kk + ((ln < 16) ? j : 2 + j);      // 0..71
    int c = 16 * nt + (ln & 15);                   // 0..127
    lds[WB2_OFF + i] = wl[r * N_OUT + c];
  }
  for (int i = tid; i < 80; i += 256) {
    lds[BA_OFF + i] = (i < 72) ? b1[i] : 0.0f;     // b1 flat [3*24] == column index
    lds[BG_OFF + i] = (i < 72) ? b2[i] : 0.0f;
  }
  for (int i = tid; i < 128; i += 256) lds[BL_OFF + i] = bl[i];
  __syncthreads();

  // per-lane bias registers
  float ba_r[5], bg_r[5], bl_r[8];
#pragma unroll
  for (int p = 0; p < 5; ++p) { ba_r[p] = lds[BA_OFF + 16 * p + lp]; bg_r[p] = lds[BG_OFF + 16 * p + lp]; }
#pragma unroll
  for (int n = 0; n < 8; ++n) bl_r[n] = lds[BL_OFF + 16 * n + lp];

  float* Sl = &lds[S_OFF + wave * S_PER_WAVE];

  const int n_tiles = n_edges >> 4;
  const int gwave   = blockIdx.x * 8 + wave;
  const int nwaves  = gridDim.x * 8;

  for (int t = gwave; t < n_tiles; t += nwaves) {
    const int e0   = t << 4;
    const int erow = e0 + lp;

    // ---- gather h_emb: lo lanes -> h[idx_s] (cols 0..11), hi lanes -> h[idx_t] (cols 12..23)
    const int idx = hi ? idx_t[erow] : idx_s[erow];
    const float* hp = h + (size_t)idx * 12;
    float xl[12], xo[12];
    *(v4f*)&xl[0] = *(const v4f*)(hp);
    *(v4f*)&xl[4] = *(const v4f*)(hp + 4);
    *(v4f*)&xl[8] = *(const v4f*)(hp + 8);
#pragma unroll
    for (int q = 0; q < 12; ++q) xo[q] = __shfl_xor(xl[q], 16, 32);

    // A fragments for GEMM1, built directly (lane's K-slice only; X never materialized)
    v2f A1[K1];
#pragma unroll
    for (int k = 0; k < 3; ++k) {
      A1[k].x     = hi ? xo[4 * k + 2] : xl[4 * k + 0];
      A1[k].y     = hi ? xo[4 * k + 3] : xl[4 * k + 1];
      A1[k + 3].x = hi ? xl[4 * k + 2] : xo[4 * k + 0];
      A1[k + 3].y = hi ? xl[4 * k + 3] : xo[4 * k + 1];
    }

    // rbf: each lane only ever needs cols ==(0,1 mod 4) [lo] or ==(2,3 mod 4) [hi]
    const float* rp = rbf + (size_t)erow * 24 + (hi ? 12 : 0);
    float rl[12], ro[12];
    *(v4f*)&rl[0] = *(const v4f*)(rp);
    *(v4f*)&rl[4] = *(const v4f*)(rp + 4);
    *(v4f*)&rl[8] = *(const v4f*)(rp + 8);
#pragma unroll
    for (int q = 0; q < 12; ++q) ro[q] = __shfl_xor(rl[q], 16, 32);
    float RR[12];                                  // RR[2m+j] = R[4m + (hi?2:0) + j]
#pragma unroll
    for (int m = 0; m < 3; ++m) {
      RR[2 * m + 0]       = hi ? ro[4 * m + 2] : rl[4 * m + 0];
      RR[2 * m + 1]       = hi ? ro[4 * m + 3] : rl[4 * m + 1];
      RR[2 * (m + 3) + 0] = hi ? rl[4 * m + 2] : ro[4 * m + 0];
      RR[2 * (m + 3) + 1] = hi ? rl[4 * m + 3] : ro[4 * m + 1];
    }

    // ---- GEMM1 + GLU, paired a/g tiles; write glu to per-wave LDS S tile ----
#pragma unroll
    for (int p = 0; p < 5; ++p) {
      v8f ca = {}; v8f cg = {};
#pragma unroll
      for (int kp = 0; kp < KP1; ++kp) {           // one b128 = two B fragments
        v4f bq = *(const v4f*)&lds[WB1_OFF + ((p * KP1 + kp) * 32 + lane) * 4];
        v2f b0 = __builtin_shufflevector(bq, bq, 0, 1);
        v2f b1v = __builtin_shufflevector(bq, bq, 2, 3);
        ca = __builtin_amdgcn_wmma_f32_16x16x4_f32(false, A1[2 * kp + 0], false, b0, (short)0, ca, false, false);
        ca = __builtin_amdgcn_wmma_f32_16x16x4_f32(false, A1[2 * kp + 1], false, b1v, (short)0, ca, false, false);
      }
#pragma unroll
      for (int kp = 0; kp < KP1; ++kp) {
        v4f bq = *(const v4f*)&lds[WB1_OFF + (((p + 5) * KP1 + kp) * 32 + lane) * 4];
        v2f b0 = __builtin_shufflevector(bq, bq, 0, 1);
        v2f b1v = __builtin_shufflevector(bq, bq, 2, 3);
        cg = __builtin_amdgcn_wmma_f32_16x16x4_f32(false, A1[2 * kp + 0], false, b0, (short)0, cg, false, false);
        cg = __builtin_amdgcn_wmma_f32_16x16x4_f32(false, A1[2 * kp + 1], false, b1v, (short)0, cg, false, false);
      }
#pragma unroll
      for (int r = 0; r < 8; ++r) {
        float a = ca[r] + ba_r[p];
        float g = cg[r] + bg_r[p];
        // fast sigmoid: v_exp_f32 + v_rcp_f32 on the TRANS pipe
        float s = __builtin_amdgcn_rcpf(1.0f + __expf(-g));
        int row = r + (hi ? 8 : 0);
        Sl[row * SSTRIDE + 16 * p + lp] = a * s;
      }
    }
    __builtin_amdgcn_wave_barrier();
    asm volatile("s_wait_dscnt 0" ::: "memory");   // LDS stores visible before transpose reads
    __builtin_amdgcn_wave_barrier();

    // ---- build GEMM2 A fragments from S (transpose via LDS), fold in rbf ----
    v2f A2[K2];
#pragma unroll
    for (int kk = 0; kk < K2; ++kk) {
      v2f sv = *(const v2f*)&Sl[lp * SSTRIDE + 4 * kk + (hi ? 2 : 0)];
      const int m6 = kk % 6;                       // compile-time after unroll
      A2[kk].x = sv.x * RR[2 * m6 + 0];
      A2[kk].y = sv.y * RR[2 * m6 + 1];
    }

    // ---- GEMM2: S[16x72] @ wl[72x128], + bl, store ----
#pragma unroll
    for (int n = 0; n < NT2; ++n) {
      v8f acc = {};
#pragma unroll
      for (int kp = 0; kp < KP2; ++kp) {           // one b128 = two B fragments
        v4f bq = *(const v4f*)&lds[WB2_OFF + ((n * KP2 + kp) * 32 + lane) * 4];
        v2f b0 = __builtin_shufflevector(bq, bq, 0, 1);
        v2f b1v = __builtin_shufflevector(bq, bq, 2, 3);
        acc = __builtin_amdgcn_wmma_f32_16x16x4_f32(false, A2[2 * kp + 0], false, b0, (short)0, acc, false, false);
        acc = __builtin_amdgcn_wmma_f32_16x16x4_f32(false, A2[2 * kp + 1], false, b1v, (short)0, acc, false, false);
      }
      float* op = out + (size_t)e0 * N_OUT + 16 * n + lp;
#pragma unroll
      for (int r = 0; r < 8; ++r) {
        int row = r + (hi ? 8 : 0);
        op[(size_t)row * N_OUT] = acc[r] + bl_r[n];
      }
    }
  }
}

extern "C" void kernel_launch(void* const* d_in, const int* in_sizes, int n_in,
                              void* d_out, int out_size, void* d_ws, size_t ws_size,
                              hipStream_t stream) {
  const float* rbf   = (const float*)d_in[0];
  const float* h     = (const float*)d_in[1];
  const int*   idx_s = (const int*)d_in[2];
  const int*   idx_t = (const int*)d_in[3];
  const float* w1    = (const float*)d_in[4];
  const float* b1    = (const float*)d_in[5];
  const float* w2    = (const float*)d_in[6];
  const float* b2    = (const float*)d_in[7];
  const float* wl    = (const float*)d_in[8];
  const float* bl    = (const float*)d_in[9];
  float* out = (float*)d_out;

  const int n_edges = in_sizes[2];                 // idx_s length == E
  const int tiles   = n_edges / 16;
  int blocks = (tiles + 7) / 8;
  if (blocks > 1024) blocks = 1024;                // grid-stride: amortize LDS weight build
  const size_t shmem = (size_t)LDS_DWORDS * sizeof(float);

  hipFuncSetAttribute((const void*)me_glu_kernel,
                      hipFuncAttributeMaxDynamicSharedMemorySize, (int)shmem);

  me_glu_kernel<<<blocks, 256, shmem, stream>>>(rbf, h, idx_s, idx_t,
                                                w1, b1, w2, b2, wl, bl,
                                                out, n_edges);
}